// GCN_19825569039034
// MI455X (gfx1250) — compile-verified
//
#include <hip/hip_runtime.h>

typedef __attribute__((ext_vector_type(2))) float v2f;
typedef __attribute__((ext_vector_type(8))) float v8f;

// ---------------- degree / normalization ----------------
__global__ void gcn_deg_kernel(const int* __restrict__ dst, float* __restrict__ deg,
                               long long ne) {
  long long e = (long long)blockIdx.x * blockDim.x + threadIdx.x;
  if (e < ne) atomicAdd(&deg[dst[e]], 1.0f);
}

__global__ void gcn_dinv_kernel(float* __restrict__ deg, int n) {
  int i = blockIdx.x * blockDim.x + threadIdx.x;
  if (i < n) deg[i] = rsqrtf(deg[i] + 1.0f);  // +1 = self loop; deg >= 1 always
}

// ---------------- fp32 WMMA GEMM with fused dinv row-scale ----------------
// out[row, :] = dinv[row] * (X[row, :] @ W)    (16 rows per wave, OC cols)
template <int OC>
__global__ __launch_bounds__(128) void gcn_gemm_wmma_scale(
    const float* __restrict__ X, const float* __restrict__ W,
    const float* __restrict__ dinv, float* __restrict__ out, int nrows, int IC) {
  const int wave = blockIdx.x * (blockDim.x >> 5) + (threadIdx.x >> 5);
  const int r0 = wave * 16;
  if (r0 >= nrows) return;  // uniform per wave -> EXEC all-ones around WMMA
  const int lane = threadIdx.x & 31;
  const int half = lane >> 4;  // 0: K+{0,1}, 1: K+{2,3}
  const int lr = lane & 15;

  constexpr int NT = OC / 16;
  v8f zero = {};
  v8f acc[NT];
#pragma unroll
  for (int t = 0; t < NT; ++t) acc[t] = zero;

  const float* xrow = X + (size_t)(r0 + lr) * IC;  // A: lane = row M
  for (int k = 0; k < IC; k += 4) {
    v2f a;
    a.x = xrow[k + 2 * half];      // VGPR0: K = k+2*half
    a.y = xrow[k + 2 * half + 1];  // VGPR1: K = k+2*half+1
#pragma unroll
    for (int t = 0; t < NT; ++t) {
      const float* wp = W + (size_t)(k + 2 * half) * OC + t * 16 + lr;  // B: lane = col N
      v2f b;
      b.x = wp[0];
      b.y = wp[OC];
      acc[t] = __builtin_amdgcn_wmma_f32_16x16x4_f32(
          false, a, false, b, (short)0, acc[t], false, false);
    }
  }
  // C/D layout: VGPR r -> row r (lanes 0-15) / row r+8 (lanes 16-31), lane = col
#pragma unroll
  for (int r = 0; r < 8; ++r) {
    const int row = r0 + r + half * 8;
    const float dv = dinv[row];
#pragma unroll
    for (int t = 0; t < NT; ++t)
      out[(size_t)row * OC + t * 16 + lr] = dv * acc[t][r];
  }
}

// ---------------- edge scatter: agg[dst] += hs[src] ----------------
template <int C>
__global__ void gcn_scatter_add(const float* __restrict__ hs, const int* __restrict__ src,
                                const int* __restrict__ dst, float* __restrict__ agg,
                                long long ne) {
  constexpr int TPE = C / 4;  // threads per edge, float4 per thread
  long long t = (long long)blockIdx.x * blockDim.x + threadIdx.x;
  long long e = t / TPE;
  int g = (int)(t % TPE);
  if (e >= ne) return;
  int s = src[e];
  int d = dst[e];
  float4 v = *reinterpret_cast<const float4*>(hs + (size_t)s * C + g * 4);
  float* p = agg + (size_t)d * C + g * 4;
  atomicAdd(p + 0, v.x);
  atomicAdd(p + 1, v.y);
  atomicAdd(p + 2, v.z);
  atomicAdd(p + 3, v.w);
}

// ---------------- finalize: agg = [relu](dinv*(agg + hs) + b) ----------------
template <int C, bool RELU>
__global__ void gcn_finalize(float* __restrict__ agg, const float* __restrict__ hs,
                             const float* __restrict__ dinv, const float* __restrict__ b,
                             int n) {
  constexpr int TPN = C / 4;
  long long t = (long long)blockIdx.x * blockDim.x + threadIdx.x;
  long long i = t / TPN;
  int g = (int)(t % TPN);
  if (i >= n) return;
  float dv = dinv[i];
  float4 a = *reinterpret_cast<const float4*>(agg + (size_t)i * C + g * 4);
  float4 h = *reinterpret_cast<const float4*>(hs + (size_t)i * C + g * 4);
  float4 bb = *reinterpret_cast<const float4*>(b + g * 4);
  float4 r;
  r.x = dv * (a.x + h.x) + bb.x;
  r.y = dv * (a.y + h.y) + bb.y;
  r.z = dv * (a.z + h.z) + bb.z;
  r.w = dv * (a.w + h.w) + bb.w;
  if (RELU) {
    r.x = fmaxf(r.x, 0.0f);
    r.y = fmaxf(r.y, 0.0f);
    r.z = fmaxf(r.z, 0.0f);
    r.w = fmaxf(r.w, 0.0f);
  }
  *reinterpret_cast<float4*>(agg + (size_t)i * C + g * 4) = r;
}

// ---------------- root selection + fused layer-2 finalize gather ----------------
__global__ void gcn_root_kernel(const float* __restrict__ x, const int* __restrict__ batch,
                                int* __restrict__ roots, int n, int ic) {
  int i = blockIdx.x * blockDim.x + threadIdx.x;
  if (i < n && x[(size_t)i * ic] == 0.0f) atomicMin(&roots[batch[i]], i);
}

__global__ void gcn_gather_out(const int* __restrict__ roots, const float* __restrict__ agg2,
                               const float* __restrict__ hs2, const float* __restrict__ dinv,
                               const float* __restrict__ b2, float* __restrict__ out,
                               int ng, int n) {
  int t = blockIdx.x * blockDim.x + threadIdx.x;
  int g = t >> 6;       // OC = 64
  int c = t & 63;
  if (g >= ng) return;
  int r = roots[g];
  float v = 0.0f;
  if (r < n)
    v = dinv[r] * (agg2[(size_t)r * 64 + c] + hs2[(size_t)r * 64 + c]) + b2[c];
  out[(size_t)g * 64 + c] = v;
}

extern "C" void kernel_launch(void* const* d_in, const int* in_sizes, int n_in,
                              void* d_out, int out_size, void* d_ws, size_t ws_size,
                              hipStream_t stream) {
  (void)n_in; (void)ws_size;
  const float* x  = (const float*)d_in[0];
  const int*   ei = (const int*)d_in[1];
  const int* batch = (const int*)d_in[2];
  // d_in[3] = num_graphs scalar (derived from out_size instead)
  const float* W1 = (const float*)d_in[4];
  const float* b1 = (const float*)d_in[5];
  const float* W2 = (const float*)d_in[6];
  const float* b2 = (const float*)d_in[7];

  const int IC = 128, HC = 128, OC = 64;
  const int n = in_sizes[0] / IC;               // 50000
  const long long ne = (long long)in_sizes[1] / 2;  // 800000
  const int ng = out_size / OC;                 // 500
  const int* srcp = ei;
  const int* dstp = ei + ne;

  // workspace layout
  char* w = (char*)d_ws;
  float* dinv = (float*)w;
  w += (((size_t)n * 4 + 255) / 256) * 256;
  float* bufA = (float*)w;                      // hs1, later hs2
  w += (size_t)n * HC * sizeof(float);
  float* bufB = (float*)w;                      // agg1, later agg2
  w += (size_t)n * HC * sizeof(float);
  int* roots = (int*)w;

  const int BT = 256;
  hipMemsetAsync(dinv, 0, (size_t)n * sizeof(float), stream);
  hipMemsetAsync(bufB, 0, (size_t)n * HC * sizeof(float), stream);
  hipMemsetAsync(roots, 0x7F, (size_t)ng * sizeof(int), stream);  // ~INT_MAX sentinel

  gcn_deg_kernel<<<(unsigned)((ne + BT - 1) / BT), BT, 0, stream>>>(dstp, dinv, ne);
  gcn_dinv_kernel<<<(n + BT - 1) / BT, BT, 0, stream>>>(dinv, n);

  const int strips = (n + 15) / 16;             // 3125, exact
  const int gblocks = (strips + 3) / 4;         // 4 waves / 128-thread block

  // layer 1: hs1 = dinv * (x @ W1)
  gcn_gemm_wmma_scale<128><<<gblocks, 128, 0, stream>>>(x, W1, dinv, bufA, n, IC);
  {
    long long tot = ne * (128 / 4);
    gcn_scatter_add<128><<<(unsigned)((tot + BT - 1) / BT), BT, 0, stream>>>(
        bufA, srcp, dstp, bufB, ne);
  }
  {
    long long tot = (long long)n * (128 / 4);
    gcn_finalize<128, true><<<(unsigned)((tot + BT - 1) / BT), BT, 0, stream>>>(
        bufB, bufA, dinv, b1, n);
  }

  // layer 2: hs2 = dinv * (agg1 @ W2)   (reuse bufA for hs2)
  gcn_gemm_wmma_scale<64><<<gblocks, 128, 0, stream>>>(bufB, W2, dinv, bufA, n, HC);
  hipMemsetAsync(bufB, 0, (size_t)n * OC * sizeof(float), stream);  // agg2
  {
    long long tot = ne * (64 / 4);
    gcn_scatter_add<64><<<(unsigned)((tot + BT - 1) / BT), BT, 0, stream>>>(
        bufA, srcp, dstp, bufB, ne);
  }

  // roots + fused finalize-at-root gather
  gcn_root_kernel<<<(n + BT - 1) / BT, BT, 0, stream>>>(x, batch, roots, n, IC);
  gcn_gather_out<<<(ng * 64 + BT - 1) / BT, BT, 0, stream>>>(
      roots, bufB, bufA, dinv, b2, (float*)d_out, ng, n);
}